// ResidualLayer_62328565399967
// MI455X (gfx1250) — compile-verified
//
#include <hip/hip_runtime.h>
#include <math.h>
#include <stdint.h>

// Problem constants (match reference)
#define N_NODES 50000
#define N_EDGES 800000
#define ETOT    (N_EDGES + N_NODES)   // self-loops appended
#define HDIM    256
#define HEADS   4
#define OUT_D   64
#define NEG_SLOPE 0.2f
#define LN_EPS  1e-5f

// LDS row stride for the A stripe: 260 floats = 1040 bytes.
//  - 16B aligned (async B128 writes legal)
//  - 4m + kb bank pattern -> conflict-free ds_load_b64 across a wave32
#define A_STRIDE 260

typedef float v2f __attribute__((ext_vector_type(2)));
typedef float v8f __attribute__((ext_vector_type(8)));

// ---------------------------------------------------------------------------
// init: amax = -inf, denom = 0, agg = 0
// ---------------------------------------------------------------------------
__global__ void init_ws_kernel(float* __restrict__ amax, float* __restrict__ denom,
                               float* __restrict__ agg) {
  int i = blockIdx.x * blockDim.x + threadIdx.x;
  if (i < N_NODES * HEADS) {
    ((unsigned int*)amax)[i] = 0xFF800000u;   // -inf
    denom[i] = 0.0f;
  }
  if (i < N_NODES * HDIM) agg[i] = 0.0f;
}

// ---------------------------------------------------------------------------
// h = x @ W (50000x256 @ 256x256), fp32 exact via V_WMMA_F32_16X16X4_F32.
// One block (8 waves) computes a full 16x256 output stripe:
//   - the 16x256 A stripe (16KB) is staged into LDS ONCE via
//     global_load_async_to_lds_b128 (+ s_wait_asynccnt), so x is read from
//     HBM exactly once.
//   - wave w owns column tiles 2w, 2w+1 (cols 32w..32w+31); A comes from LDS
//     (ds_load_b64, conflict-free via A_STRIDE padding), B (L2-resident 256KB)
//     from global.
// A layout (16x4 f32):  lane m=lane&15, elem j holds K = 2*(lane>>4)+j
// B layout (4x16 f32):  lane n=lane&15, elem j holds K = 2*(lane>>4)+j
// C layout (16x16 f32): VGPR r: lanes0-15 -> M=r, lanes16-31 -> M=r+8
// ---------------------------------------------------------------------------
__global__ __launch_bounds__(256) void gemm_h_kernel(const float* __restrict__ x,
                                                     const float* __restrict__ W,
                                                     float* __restrict__ h) {
  __shared__ float As[16 * A_STRIDE];

  const int tid  = threadIdx.x;
  const int lane = tid & 31;
  const int wave = tid >> 5;
  const int row0 = blockIdx.x * 16;          // 3125 row stripes

  // ---- async stage A stripe: 16 rows x 256 f32 = 1024 x 16B chunks -------
  {
    const unsigned ldsBase = (unsigned)(uintptr_t)&As[0];   // LDS byte offset
#pragma unroll
    for (int it = 0; it < 4; ++it) {
      const int c  = tid + it * 256;                        // chunk 0..1023
      const int r  = c >> 6;                                // row 0..15
      const int cf = (c & 63) << 2;                         // col in floats
      const unsigned long long gp =
          (unsigned long long)(uintptr_t)(x + (size_t)(row0 + r) * HDIM + cf);
      const unsigned ldsAddr = ldsBase + (unsigned)(r * (A_STRIDE * 4) + ((c & 63) << 4));
      asm volatile("global_load_async_to_lds_b128 %0, %1, off"
                   :: "v"(ldsAddr), "v"(gp) : "memory");
    }
    asm volatile("s_wait_asynccnt 0x0" ::: "memory");
  }
  __syncthreads();

  // ---- WMMA main loop -----------------------------------------------------
  const int m     = lane & 15;
  const int khalf = (lane >> 4) << 1;        // 0 or 2
  const int col0  = wave * 32;               // 2 column tiles per wave

  v8f acc0 = {0.f, 0.f, 0.f, 0.f, 0.f, 0.f, 0.f, 0.f};
  v8f acc1 = {0.f, 0.f, 0.f, 0.f, 0.f, 0.f, 0.f, 0.f};

  const float* arow = &As[m * A_STRIDE];
  for (int k = 0; k < HDIM; k += 4) {
    const int kb = k + khalf;
    v2f a = *(const v2f*)(arow + kb);                        // ds_load_b64
    v2f b0, b1;
    b0[0] = W[(size_t)kb       * HDIM + col0 + m];
    b0[1] = W[(size_t)(kb + 1) * HDIM + col0 + m];
    b1[0] = W[(size_t)kb       * HDIM + col0 + 16 + m];
    b1[1] = W[(size_t)(kb + 1) * HDIM + col0 + 16 + m];
    acc0 = __builtin_amdgcn_wmma_f32_16x16x4_f32(false, a, false, b0,
                                                 (short)0, acc0, false, false);
    acc1 = __builtin_amdgcn_wmma_f32_16x16x4_f32(false, a, false, b1,
                                                 (short)0, acc1, false, false);
  }

  const int rbase = (lane >> 4) ? 8 : 0;
#pragma unroll
  for (int r = 0; r < 8; ++r) {
    h[(size_t)(row0 + rbase + r) * HDIM + col0 + m]      = acc0[r];
    h[(size_t)(row0 + rbase + r) * HDIM + col0 + 16 + m] = acc1[r];
  }
}

// ---------------------------------------------------------------------------
// a_s[n,hd] = <h[n,hd,:], att_src[hd,:]>,  a_d likewise. Thread per (n, head).
// ---------------------------------------------------------------------------
__global__ void node_attn_kernel(const float* __restrict__ h,
                                 const float* __restrict__ att_src,
                                 const float* __restrict__ att_dst,
                                 float* __restrict__ a_s, float* __restrict__ a_d) {
  int i = blockIdx.x * blockDim.x + threadIdx.x;
  if (i >= N_NODES * HEADS) return;
  const int n = i >> 2, hd = i & 3;
  const float* hp = h + (size_t)n * HDIM + hd * OUT_D;
  const float* as = att_src + hd * OUT_D;
  const float* ad = att_dst + hd * OUT_D;
  float s = 0.f, d = 0.f;
#pragma unroll 4
  for (int j = 0; j < OUT_D; j += 4) {
    float4 hv = *(const float4*)(hp + j);
    float4 av = *(const float4*)(as + j);
    float4 dv = *(const float4*)(ad + j);
    s += hv.x * av.x + hv.y * av.y + hv.z * av.z + hv.w * av.w;
    d += hv.x * dv.x + hv.y * dv.y + hv.z * dv.z + hv.w * dv.w;
  }
  a_s[i] = s;
  a_d[i] = d;
}

// signed/unsigned bit-trick float atomic max (amax initialized to -inf bits)
__device__ __forceinline__ void atomicMaxF(float* addr, float v) {
  if (v >= 0.0f) atomicMax((int*)addr, __float_as_int(v));
  else           atomicMin((unsigned int*)addr, __float_as_uint(v));
}

// ---------------------------------------------------------------------------
// alpha[e,hd] = leakyrelu(a_s[src]+a_d[dst]); amax[dst,hd] = max(...)
// Thread per (edge, head). Self-loops for e >= N_EDGES.
// ---------------------------------------------------------------------------
__global__ void edge_alpha_kernel(const int* __restrict__ ei,
                                  const float* __restrict__ a_s,
                                  const float* __restrict__ a_d,
                                  float* __restrict__ alpha,
                                  float* __restrict__ amax) {
  long long t = (long long)blockIdx.x * blockDim.x + threadIdx.x;
  if (t >= (long long)ETOT * HEADS) return;
  const int e = (int)(t >> 2), hd = (int)(t & 3);
  const int s = (e < N_EDGES) ? ei[e]            : (e - N_EDGES);
  const int d = (e < N_EDGES) ? ei[N_EDGES + e]  : (e - N_EDGES);
  float v = a_s[s * HEADS + hd] + a_d[d * HEADS + hd];
  v = (v > 0.f) ? v : NEG_SLOPE * v;
  alpha[t] = v;
  atomicMaxF(&amax[d * HEADS + hd], v);
}

// ---------------------------------------------------------------------------
// ex[e,hd] = exp(alpha - amax[dst]); denom[dst,hd] += ex (in-place over alpha)
// ---------------------------------------------------------------------------
__global__ void edge_exp_kernel(const int* __restrict__ ei,
                                const float* __restrict__ amax,
                                float* __restrict__ alpha,
                                float* __restrict__ denom) {
  long long t = (long long)blockIdx.x * blockDim.x + threadIdx.x;
  if (t >= (long long)ETOT * HEADS) return;
  const int e = (int)(t >> 2), hd = (int)(t & 3);
  const int d = (e < N_EDGES) ? ei[N_EDGES + e] : (e - N_EDGES);
  float ex = expf(alpha[t] - amax[d * HEADS + hd]);
  alpha[t] = ex;
  atomicAdd(&denom[d * HEADS + hd], ex);
}

// ---------------------------------------------------------------------------
// agg[dst,:] += (ex/denom[dst]) * h[src,:]   — one wave32 per edge,
// 8 floats/lane via two B128 loads, 8 global f32 atomics/lane (L2-resident).
// ---------------------------------------------------------------------------
__global__ __launch_bounds__(256) void edge_agg_kernel(const int* __restrict__ ei,
                                                       const float* __restrict__ ex,
                                                       const float* __restrict__ denom,
                                                       const float* __restrict__ h,
                                                       float* __restrict__ agg) {
  const int lane = threadIdx.x & 31;
  const long long e = (long long)blockIdx.x * 8 + (threadIdx.x >> 5);
  if (e >= ETOT) return;
  const int ie = (int)e;
  const int s = (ie < N_EDGES) ? ei[ie]           : (ie - N_EDGES);
  const int d = (ie < N_EDGES) ? ei[N_EDGES + ie] : (ie - N_EDGES);
  const int hd = lane >> 3;                         // 8 lanes per head
  const float c = ex[(size_t)ie * HEADS + hd] / denom[d * HEADS + hd];

  const float* hp = h + (size_t)s * HDIM + lane * 8;
  float4 h0 = *(const float4*)(hp);
  float4 h1 = *(const float4*)(hp + 4);
  float* ap = agg + (size_t)d * HDIM + lane * 8;
  atomicAdd(ap + 0, c * h0.x);
  atomicAdd(ap + 1, c * h0.y);
  atomicAdd(ap + 2, c * h0.z);
  atomicAdd(ap + 3, c * h0.w);
  atomicAdd(ap + 4, c * h1.x);
  atomicAdd(ap + 5, c * h1.y);
  atomicAdd(ap + 6, c * h1.z);
  atomicAdd(ap + 7, c * h1.w);
}

// ---------------------------------------------------------------------------
// out = LayerNorm(relu(agg + bias) + x) * gamma + beta — one wave32 per node,
// wave-wide shuffle butterflies for mean/var.
// ---------------------------------------------------------------------------
__global__ __launch_bounds__(256) void finalize_kernel(const float* __restrict__ agg,
                                                       const float* __restrict__ x,
                                                       const float* __restrict__ bias,
                                                       const float* __restrict__ gamma,
                                                       const float* __restrict__ beta,
                                                       float* __restrict__ out) {
  const int lane = threadIdx.x & 31;
  const int n = blockIdx.x * 8 + (threadIdx.x >> 5);
  if (n >= N_NODES) return;
  const int c0 = lane * 8;
  const float* ap = agg + (size_t)n * HDIM + c0;
  const float* xp = x + (size_t)n * HDIM + c0;

  float y[8];
  float sum = 0.f, ss = 0.f;
#pragma unroll
  for (int j = 0; j < 8; ++j) {
    float v = ap[j] + bias[c0 + j];
    v = (v > 0.f) ? v : 0.f;        // relu
    v += xp[j];                     // residual
    y[j] = v;
    sum += v;
    ss += v * v;
  }
#pragma unroll
  for (int off = 16; off >= 1; off >>= 1) {
    sum += __shfl_xor(sum, off, 32);
    ss  += __shfl_xor(ss,  off, 32);
  }
  const float mu   = sum * (1.0f / HDIM);
  const float var  = ss * (1.0f / HDIM) - mu * mu;
  const float rstd = rsqrtf(var + LN_EPS);

  float* op = out + (size_t)n * HDIM + c0;
#pragma unroll
  for (int j = 0; j < 8; ++j)
    op[j] = (y[j] - mu) * rstd * gamma[c0 + j] + beta[c0 + j];
}

// ---------------------------------------------------------------------------
extern "C" void kernel_launch(void* const* d_in, const int* in_sizes, int n_in,
                              void* d_out, int out_size, void* d_ws, size_t ws_size,
                              hipStream_t stream) {
  const float* x       = (const float*)d_in[0];
  const int*   ei      = (const int*)d_in[1];   // [2, E] flat: src then dst
  const float* W       = (const float*)d_in[2];
  const float* att_src = (const float*)d_in[3];
  const float* att_dst = (const float*)d_in[4];
  const float* bias    = (const float*)d_in[5];
  const float* gamma   = (const float*)d_in[6];
  const float* beta    = (const float*)d_in[7];
  float* out = (float*)d_out;

  // Workspace carve-out (floats)
  float* ws    = (float*)d_ws;
  float* h     = ws;                                  // N*H
  float* agg   = h + (size_t)N_NODES * HDIM;          // N*H
  float* a_s   = agg + (size_t)N_NODES * HDIM;        // N*HEADS
  float* a_d   = a_s + (size_t)N_NODES * HEADS;       // N*HEADS
  float* amax  = a_d + (size_t)N_NODES * HEADS;       // N*HEADS
  float* denom = amax + (size_t)N_NODES * HEADS;      // N*HEADS
  float* alpha = denom + (size_t)N_NODES * HEADS;     // ETOT*HEADS (alpha->ex)

  // 1) init accumulators
  init_ws_kernel<<<(N_NODES * HDIM + 255) / 256, 256, 0, stream>>>(amax, denom, agg);

  // 2) h = x @ W  (WMMA fp32, async-LDS staged A)
  gemm_h_kernel<<<N_NODES / 16, 256, 0, stream>>>(x, W, h);

  // 3) per-node attention scalars
  node_attn_kernel<<<(N_NODES * HEADS + 255) / 256, 256, 0, stream>>>(
      h, att_src, att_dst, a_s, a_d);

  // 4) edge logits + segment max
  const long long tEH = (long long)ETOT * HEADS;
  edge_alpha_kernel<<<(unsigned)((tEH + 255) / 256), 256, 0, stream>>>(
      ei, a_s, a_d, alpha, amax);

  // 5) exp + segment sum
  edge_exp_kernel<<<(unsigned)((tEH + 255) / 256), 256, 0, stream>>>(
      ei, amax, alpha, denom);

  // 6) weighted scatter-add aggregation (one wave per edge)
  edge_agg_kernel<<<(ETOT + 7) / 8, 256, 0, stream>>>(ei, alpha, denom, h, agg);

  // 7) bias + relu + residual + layernorm
  finalize_kernel<<<(N_NODES + 7) / 8, 256, 0, stream>>>(agg, x, bias, gamma, beta, out);
}